// LightConv_661424963755
// MI455X (gfx1250) — compile-verified
//
#include <hip/hip_runtime.h>

typedef __attribute__((ext_vector_type(2))) float v2f;
typedef __attribute__((ext_vector_type(8))) float v8f;

// ---------------------------------------------------------------------------
// Pass 1: per-edge degree accumulation (L2-native f32 atomics).
// ---------------------------------------------------------------------------
__global__ void lc_degree_kernel(const int* __restrict__ src,
                                 const int* __restrict__ dst,
                                 float* __restrict__ deg_out,
                                 float* __restrict__ deg_in,
                                 int n_edges) {
    int e = blockIdx.x * blockDim.x + threadIdx.x;
    if (e < n_edges) {
        atomicAdd(&deg_out[src[e]], 1.0f);
        atomicAdd(&deg_in[dst[e]], 1.0f);
    }
}

// ---------------------------------------------------------------------------
// Pass 2: deg -> clip(deg,1)^-0.5 in place (covers both deg arrays at once).
// ---------------------------------------------------------------------------
__global__ void lc_rsqrt_kernel(float* __restrict__ deg, int n2) {
    int i = blockIdx.x * blockDim.x + threadIdx.x;
    if (i < n2) {
        float v = deg[i];
        v = (v < 1.0f) ? 1.0f : v;
        deg[i] = 1.0f / sqrtf(v);
    }
}

// ---------------------------------------------------------------------------
// Pass 3: edge-parallel SpMM. 16 lanes own one edge; each lane moves a float4
// (lane group reads one contiguous 256B feature row -> global_load_b128,
// then 4 global_atomic_add_f32 into the contiguous destination row).
// Pre-scaling by inv_out[src] is fused into the gather.
// ---------------------------------------------------------------------------
__global__ void lc_spmm_kernel(const float* __restrict__ feat,
                               const int* __restrict__ src,
                               const int* __restrict__ dst,
                               const float* __restrict__ inv_out,
                               float* __restrict__ out,
                               int n_edges) {
    int e = blockIdx.x * (blockDim.x >> 4) + (threadIdx.x >> 4);
    int c = threadIdx.x & 15;              // float4 slot within the 64-wide row
    if (e >= n_edges) return;

    int s = src[e];                        // uniform across the 16-lane group
    int d = dst[e];
    float sc = inv_out[s];

    const float4 f =
        *reinterpret_cast<const float4*>(feat + (size_t)s * 64 + (size_t)c * 4);

    float* o = out + (size_t)d * 64 + (size_t)c * 4;
    atomicAdd(o + 0, sc * f.x);
    atomicAdd(o + 1, sc * f.y);
    atomicAdd(o + 2, sc * f.z);
    atomicAdd(o + 3, sc * f.w);
}

// ---------------------------------------------------------------------------
// Pass 4: out_tile(16x16) = diag(inv_in) * agg_tile via V_WMMA_F32_16X16X4_F32.
// One wave32 per (16-node x 16-feature) tile; 4 K-slices of the 16x16 diagonal
// A-matrix accumulate the exact row-scaled result in the f32 C/D registers.
//   A 16x4 slice j : lane (m = lane&15, half = lane>>4)
//       a0 holds A[m][4j+2*half], a1 holds A[m][4j+2*half+1]
//   B 4x16 slice j : lane n = lane&15
//       b0 holds B[4j+2*half][n], b1 holds B[4j+2*half+1][n]
//   D 16x16        : vgpr r holds D[r + 8*half][lane&15]
// Full tiles take a wave-uniform unguarded fast path (no EXEC diamonds per
// load); only the ragged tail tile (never hit for n_nodes % 16 == 0) pays for
// per-lane bounds checks.
// ---------------------------------------------------------------------------
__global__ void lc_scale_wmma_kernel(float* __restrict__ out,
                                     const float* __restrict__ inv_in,
                                     int n_nodes, int total_tiles) {
    int wave = blockIdx.x * (blockDim.x >> 5) + (threadIdx.x >> 5);
    if (wave >= total_tiles) return;       // wave-uniform exit: EXEC stays all-1s

    const int tile_m = wave >> 2;          // 64 feats -> 4 feature tiles
    const int tile_n = wave & 3;
    const int lane   = threadIdx.x & 31;
    const int m      = lane & 15;
    const int half   = lane >> 4;
    const int node_base = tile_m << 4;
    const int feat_base = tile_n << 4;

    v8f c = {};

    if (node_base + 16 <= n_nodes) {
        // -------- fast path: full tile, no per-lane guards --------
        const float s = inv_in[node_base + m];
        const float* bp = out + (size_t)node_base * 64 + feat_base + m;
#pragma unroll
        for (int j = 0; j < 4; ++j) {
            const int k0 = 4 * j + 2 * half;
            v2f a;
            a[0] = (m == k0)     ? s : 0.0f;
            a[1] = (m == k0 + 1) ? s : 0.0f;
            v2f b;
            b[0] = bp[(size_t)k0 * 64];
            b[1] = bp[(size_t)(k0 + 1) * 64];
            c = __builtin_amdgcn_wmma_f32_16x16x4_f32(
                    false, a, false, b, (short)0, c, false, false);
        }
        float* dp = out + (size_t)(node_base + 8 * half) * 64 + feat_base + m;
#pragma unroll
        for (int r = 0; r < 8; ++r)
            dp[(size_t)r * 64] = c[r];
    } else {
        // -------- tail path: ragged tile, per-lane guards --------
        float s = 0.0f;
        if (node_base + m < n_nodes) s = inv_in[node_base + m];
#pragma unroll
        for (int j = 0; j < 4; ++j) {
            const int k0 = 4 * j + 2 * half;
            v2f a;
            a[0] = (m == k0)     ? s : 0.0f;
            a[1] = (m == k0 + 1) ? s : 0.0f;
            const int r0 = node_base + k0;
            v2f b;
            b[0] = (r0 < n_nodes)
                     ? out[(size_t)r0 * 64 + feat_base + m] : 0.0f;
            b[1] = (r0 + 1 < n_nodes)
                     ? out[(size_t)(r0 + 1) * 64 + feat_base + m] : 0.0f;
            c = __builtin_amdgcn_wmma_f32_16x16x4_f32(
                    false, a, false, b, (short)0, c, false, false);
        }
#pragma unroll
        for (int r = 0; r < 8; ++r) {
            const int row = node_base + r + 8 * half;
            if (row < n_nodes)
                out[(size_t)row * 64 + feat_base + m] = c[r];
        }
    }
}

// ---------------------------------------------------------------------------
// Host-side orchestration (graph-capture safe: async memsets + kernels only).
// ---------------------------------------------------------------------------
extern "C" void kernel_launch(void* const* d_in, const int* in_sizes, int n_in,
                              void* d_out, int out_size, void* d_ws, size_t ws_size,
                              hipStream_t stream) {
    const float* features = (const float*)d_in[0];
    const int*   src      = (const int*)d_in[1];
    const int*   dst      = (const int*)d_in[2];
    float*       out      = (float*)d_out;

    const int n_nodes = in_sizes[0] / 64;
    const int n_edges = in_sizes[1];

    float* deg_out = (float*)d_ws;            // n_nodes floats
    float* deg_in  = deg_out + n_nodes;       // n_nodes floats

    // Zero accumulator + degree scratch (capture-safe).
    hipMemsetAsync(d_out, 0, (size_t)out_size * sizeof(float), stream);
    hipMemsetAsync(d_ws, 0, (size_t)(2 * n_nodes) * sizeof(float), stream);

    // Pass 1: degrees.
    {
        int threads = 256;
        int blocks  = (n_edges + threads - 1) / threads;
        lc_degree_kernel<<<blocks, threads, 0, stream>>>(src, dst, deg_out,
                                                         deg_in, n_edges);
    }

    // Pass 2: clip + rsqrt over both degree arrays.
    {
        int n2 = 2 * n_nodes;
        int threads = 256;
        int blocks  = (n2 + threads - 1) / threads;
        lc_rsqrt_kernel<<<blocks, threads, 0, stream>>>(deg_out, n2);
    }

    // Pass 3: SpMM scatter-add (16 edges per 256-thread block).
    {
        int threads = 256;
        int edges_per_block = threads / 16;
        int blocks = (n_edges + edges_per_block - 1) / edges_per_block;
        lc_spmm_kernel<<<blocks, threads, 0, stream>>>(features, src, dst,
                                                       deg_out, out, n_edges);
    }

    // Pass 4: WMMA row scaling by inv_in (one wave per 16x16 tile).
    {
        int tiles_m = (n_nodes + 15) / 16;
        int total_tiles = tiles_m * 4;        // D=64 -> 4 feature tiles
        int threads = 256;                    // 8 waves per block
        int waves_per_block = threads / 32;
        int blocks = (total_tiles + waves_per_block - 1) / waves_per_block;
        lc_scale_wmma_kernel<<<blocks, threads, 0, stream>>>(out, deg_in,
                                                             n_nodes,
                                                             total_tiles);
    }
}